// GraphFlasback_12043088298507
// MI455X (gfx1250) — compile-verified
//
#include <hip/hip_runtime.h>
#include <hip/hip_bf16.h>
#include <math.h>

// ---------------- problem constants (match reference) ----------------
#define S_LEN 100
#define B_SZ  1024
#define P_SZ  50000
#define U_SZ  20000
#define H_SZ  128
#define DT_DEG 16
#define DI_DEG 32
#define LT_INV (1.0f / 1000.0f)
#define LS_INV (1.0f / 1.0f)

typedef float v2f __attribute__((ext_vector_type(2)));
typedef float v8f __attribute__((ext_vector_type(8)));

// Async global->LDS path (CDNA5): guarded so the file compiles on any toolchain.
#if defined(__has_builtin)
#if __has_builtin(__builtin_amdgcn_global_load_async_to_lds_b64)
#define HAVE_ASYNC_LDS 1
#endif
#endif

#if defined(HAVE_ASYNC_LDS)
// builtin parameter type per hipcc diagnostic:
//   'int __attribute__((vector_size(2*sizeof(int)))) __device__ *'
typedef int v2i_b64 __attribute__((vector_size(8)));
typedef __attribute__((address_space(1))) v2i_b64* async_gptr;
typedef __attribute__((address_space(3))) v2i_b64* async_lptr;
#endif

__device__ __forceinline__ void wait_asynccnt0() {
#if defined(__has_builtin)
#if __has_builtin(__builtin_amdgcn_s_wait_asynccnt)
  __builtin_amdgcn_s_wait_asynccnt(0);
  return;
#else
  asm volatile("s_wait_asynccnt 0" ::: "memory");
  return;
#endif
#else
  asm volatile("s_wait_asynccnt 0" ::: "memory");
#endif
}

// D = A(16xK) * W^T(Kx16) + C via V_WMMA_F32_16X16X4_F32.
// Lane mapping per CDNA5 ISA 7.12.2 (32-bit A 16x4 / B 4x16 / C 16x16 layouts):
//   lane l: r = l&15, half = l>>4; A frag = A[r][k+2*half .. +1],
//   B frag = W[r][k+2*half .. +1], D elem v: row = v + 8*half, col = r.

// ---------------------------------------------------------------------
// Kernel 1: x_emb[s,b,:] = sum_j trans_vals[x*16+j] * enc[trans_cols[x*16+j], :]
// ---------------------------------------------------------------------
__global__ void k_xemb(const int* __restrict__ x, const float* __restrict__ enc,
                       const int* __restrict__ tcols, const float* __restrict__ tvals,
                       float* __restrict__ x_emb) {
  const int wave = blockIdx.x * (blockDim.x >> 5) + (threadIdx.x >> 5);
  const int lane = threadIdx.x & 31;
  const int poi = x[wave];
  const int base = poi * DT_DEG;
  float a0 = 0.f, a1 = 0.f, a2 = 0.f, a3 = 0.f;
#pragma unroll
  for (int j = 0; j < DT_DEG; ++j) {
    const int c = tcols[base + j];
    const float v = tvals[base + j];
    const float4 e = ((const float4*)(enc + (size_t)c * H_SZ))[lane];
    a0 += v * e.x; a1 += v * e.y; a2 += v * e.z; a3 += v * e.w;
  }
  float4 o; o.x = a0; o.y = a1; o.z = a2; o.w = a3;
  ((float4*)(x_emb + (size_t)wave * H_SZ))[lane] = o;
}

// ---------------------------------------------------------------------
// Kernel 2: user_pref[b,:] for the 1024 active users only
// ---------------------------------------------------------------------
__global__ void k_upref(const int* __restrict__ active_user, const float* __restrict__ enc,
                        const int* __restrict__ icols, const float* __restrict__ ivals,
                        float* __restrict__ upref) {
  const int b = blockIdx.x * (blockDim.x >> 5) + (threadIdx.x >> 5);
  const int lane = threadIdx.x & 31;
  const int u = active_user[b];
  const int base = u * DI_DEG;
  float a0 = 0.f, a1 = 0.f, a2 = 0.f, a3 = 0.f;
#pragma unroll
  for (int j = 0; j < DI_DEG; ++j) {
    const int c = icols[base + j];
    const float v = ivals[base + j];
    const float4 e = ((const float4*)(enc + (size_t)c * H_SZ))[lane];
    a0 += v * e.x; a1 += v * e.y; a2 += v * e.z; a3 += v * e.w;
  }
  float4 o; o.x = a0; o.y = a1; o.z = a2; o.w = a3;
  ((float4*)(upref + (size_t)b * H_SZ))[lane] = o;
}

// ---------------------------------------------------------------------
// Kernel 3: attention weights w[b,s]
// ---------------------------------------------------------------------
__global__ void k_weights(const float* __restrict__ x_emb, const float* __restrict__ upref,
                          const float* __restrict__ t, const float* __restrict__ sxy,
                          const int* __restrict__ length, float* __restrict__ wbuf) {
  const int wave = blockIdx.x * (blockDim.x >> 5) + (threadIdx.x >> 5);
  const int lane = threadIdx.x & 31;
  const int ss = wave / B_SZ;
  const int b  = wave % B_SZ;
  const float4 up = ((const float4*)(upref + (size_t)b * H_SZ))[lane];
  const float4 xe = ((const float4*)(x_emb + (size_t)wave * H_SZ))[lane];
  float dx = up.x - xe.x, dy = up.y - xe.y, dz = up.z - xe.z, dw = up.w - xe.w;
  float sum = dx * dx + dy * dy + dz * dz + dw * dw;
#pragma unroll
  for (int m = 16; m >= 1; m >>= 1) sum += __shfl_xor(sum, m, 32);
  const float uls = __expf(-sqrtf(sum));
  const int L = length[b];
  float wv = 0.f;
  if (ss < L) {
    const int idx = L - 1;
    const float dist_t = t[(size_t)idx * B_SZ + b] - t[(size_t)ss * B_SZ + b];
    const float slx = sxy[((size_t)idx * B_SZ + b) * 2 + 0];
    const float sly = sxy[((size_t)idx * B_SZ + b) * 2 + 1];
    const float scx = sxy[((size_t)ss * B_SZ + b) * 2 + 0];
    const float scy = sxy[((size_t)ss * B_SZ + b) * 2 + 1];
    const float ex = slx - scx, ey = sly - scy;
    const float dist_s = sqrtf(ex * ex + ey * ey);
    wv = __expf(-dist_t * LT_INV) * __expf(-dist_s * LS_INV) * uls;
  }
  if (lane == 0) wbuf[(size_t)b * S_LEN + ss] = wv;
}

// ---------------------------------------------------------------------
// Kernel 4: z = x_emb @ W_ih^T + b_ih + b_hh   (M = S*B, N = K = 128)
// 16x128 per wave. All 9 fragments of a k-step loaded into distinct locals
// BEFORE the 8 independent WMMAs, unroll x2 -> many loads in flight.
// ---------------------------------------------------------------------
__global__ void k_zgemm(const float* __restrict__ x_emb, const float* __restrict__ W_ih,
                        const float* __restrict__ b_ih, const float* __restrict__ b_hh,
                        float* __restrict__ z) {
  const int wave = blockIdx.x * (blockDim.x >> 5) + (threadIdx.x >> 5);
  const int m0 = wave << 4;
  const int lane = threadIdx.x & 31;
  const int half = lane >> 4;
  const int r = lane & 15;
  const float* arow = x_emb + (size_t)(m0 + r) * H_SZ + (half << 1);
  const float* wbase = W_ih + (size_t)r * H_SZ + (half << 1);
  v8f zero = {};
  v8f acc[8];
#pragma unroll
  for (int j = 0; j < 8; ++j) acc[j] = zero;
#pragma unroll 2
  for (int k = 0; k < H_SZ; k += 4) {
    v2f a, b[8];
    a.x = arow[k]; a.y = arow[k + 1];
#pragma unroll
    for (int j = 0; j < 8; ++j) {
      b[j].x = wbase[j * 16 * H_SZ + k];
      b[j].y = wbase[j * 16 * H_SZ + k + 1];
    }
#pragma unroll
    for (int j = 0; j < 8; ++j)
      acc[j] = __builtin_amdgcn_wmma_f32_16x16x4_f32(false, a, false, b[j],
                                                     (short)0, acc[j], false, false);
  }
#pragma unroll
  for (int j = 0; j < 8; ++j) {
    const float bias = b_ih[j * 16 + r] + b_hh[j * 16 + r];
#pragma unroll
    for (int v = 0; v < 8; ++v) {
      z[(size_t)(m0 + v + half * 8) * H_SZ + j * 16 + r] = acc[j][v] + bias;
    }
  }
}

// ---------------------------------------------------------------------
// Kernel 5: one RNN step (launched S times, in-place on z):
//   z[s] <- tanh(z[s] + h_prev @ W_hh^T)     (M = B = 1024, N = K = 128)
// ---------------------------------------------------------------------
__global__ void k_rnn_step(const float* __restrict__ hprev, const float* __restrict__ W_hh,
                           float* __restrict__ zs) {
  const int wave = blockIdx.x * (blockDim.x >> 5) + (threadIdx.x >> 5);
  const int mt = wave >> 3;
  const int nt = wave & 7;
  const int m0 = mt << 4, n0 = nt << 4;
  const int lane = threadIdx.x & 31;
  const int half = lane >> 4;
  const int r = lane & 15;
  const float* arow = hprev + (size_t)(m0 + r) * H_SZ + (half << 1);
  const float* wrow = W_hh + (size_t)(n0 + r) * H_SZ + (half << 1);
  v8f acc = {};
#pragma unroll
  for (int k = 0; k < H_SZ; k += 4) {
    v2f a, b;
    a.x = arow[k]; a.y = arow[k + 1];
    b.x = wrow[k]; b.y = wrow[k + 1];
    acc = __builtin_amdgcn_wmma_f32_16x16x4_f32(false, a, false, b,
                                                (short)0, acc, false, false);
  }
#pragma unroll
  for (int v = 0; v < 8; ++v) {
    const int row = m0 + v + half * 8;
    const size_t off = (size_t)row * H_SZ + n0 + r;
    zs[off] = tanhf(acc[v] + zs[off]);
  }
}

// ---------------------------------------------------------------------
// Kernel 6: flashback pooling + concat with user embedding
// ---------------------------------------------------------------------
__global__ void k_pool(const float* __restrict__ hseq, const float* __restrict__ wbuf,
                       const float* __restrict__ user_enc, const int* __restrict__ active_user,
                       float* __restrict__ cat) {
  const int b = blockIdx.x;
  const int hIdx = threadIdx.x;
  float num = 0.f, den = 0.f;
  for (int ss = 0; ss < S_LEN; ++ss) {
    const float wv = wbuf[(size_t)b * S_LEN + ss];
    den += wv;
    num += wv * hseq[((size_t)ss * B_SZ + b) * H_SZ + hIdx];
  }
  cat[(size_t)b * (2 * H_SZ) + hIdx] = num / den;
  cat[(size_t)b * (2 * H_SZ) + H_SZ + hIdx] =
      user_enc[(size_t)active_user[b] * H_SZ + hIdx];
}

// ---------------------------------------------------------------------
// Kernel 7: final FC  out = cat @ fc_W^T + fc_b   (M=1024, N=50000, K=256)
// Block = 8 waves = 512M x 80N. fc_W tile staged in LDS (async global->LDS
// when available) in two K=128 chunks (80 x 132 floats = 42KB, padded stride
// for bank-conflict-free ds_load_b64 fragments). Each wave: 64x80 register
// tile = 20 f32-WMMA accumulators; A frags from global (L2), B frags from LDS.
// fc_W L2 traffic: 819MB -> ~102MB.
// ---------------------------------------------------------------------
#define FC_NB   80                 // N per block
#define FC_KC   128                // K chunk
#define FC_LDST (FC_KC + 4)        // padded LDS row stride (floats)

__global__ void k_fc(const float* __restrict__ cat, const float* __restrict__ fc_W,
                     const float* __restrict__ fc_b, float* __restrict__ out) {
  __shared__ float ldsB[FC_NB * FC_LDST];   // 42,240 B
  const int NG = P_SZ / FC_NB;              // 625 N-groups
  const int mblk = blockIdx.x / NG;         // 2 M-blocks (512 rows each)
  const int ng   = blockIdx.x % NG;
  const int n0 = ng * FC_NB;
  const int widx = threadIdx.x >> 5;        // 0..7
  const int m0 = mblk * 512 + widx * 64;
  const int lane = threadIdx.x & 31;
  const int half = lane >> 4;
  const int r = lane & 15;
  const int K2 = 2 * H_SZ;                  // 256
  const int tid = threadIdx.x;

  const float* arow0 = cat + (size_t)(m0 + r) * K2 + (half << 1);
  const float* brow0 = ldsB + r * FC_LDST + (half << 1);

  v8f zero = {};
  v8f acc[4][5];
#pragma unroll
  for (int i = 0; i < 4; ++i)
#pragma unroll
    for (int j = 0; j < 5; ++j) acc[i][j] = zero;

  for (int kc = 0; kc < K2; kc += FC_KC) {
    __syncthreads();   // protect LDS from previous chunk's readers
    // stage fc_W[n0 .. n0+80), cols [kc, kc+128) : 80 rows x 64 float2
#pragma unroll
    for (int i = 0; i < 20; ++i) {
      const int c = tid + i * 256;          // 0..5119
      const int row = c >> 6;
      const int col2 = (c & 63) << 1;
      const float* g = fc_W + (size_t)(n0 + row) * K2 + kc + col2;
      float* l = ldsB + row * FC_LDST + col2;
#if defined(HAVE_ASYNC_LDS)
      __builtin_amdgcn_global_load_async_to_lds_b64(
          (async_gptr)g, (async_lptr)l, 0, 0);
#else
      const float2 v = *(const float2*)g;
      *(float2*)l = v;
#endif
    }
#if defined(HAVE_ASYNC_LDS)
    wait_asynccnt0();
#endif
    __syncthreads();

#pragma unroll 2
    for (int kk = 0; kk < FC_KC; kk += 4) {
      const int k = kc + kk;
      v2f a[4], b[5];
#pragma unroll
      for (int i = 0; i < 4; ++i) {
        a[i].x = arow0[(size_t)i * 16 * K2 + k];
        a[i].y = arow0[(size_t)i * 16 * K2 + k + 1];
      }
#pragma unroll
      for (int j = 0; j < 5; ++j) {
        b[j].x = brow0[j * 16 * FC_LDST + kk];
        b[j].y = brow0[j * 16 * FC_LDST + kk + 1];
      }
#pragma unroll
      for (int i = 0; i < 4; ++i)
#pragma unroll
        for (int j = 0; j < 5; ++j)
          acc[i][j] = __builtin_amdgcn_wmma_f32_16x16x4_f32(
              false, a[i], false, b[j], (short)0, acc[i][j], false, false);
    }
  }

#pragma unroll
  for (int j = 0; j < 5; ++j) {
    const float bias = fc_b[n0 + j * 16 + r];
#pragma unroll
    for (int i = 0; i < 4; ++i) {
#pragma unroll
      for (int v = 0; v < 8; ++v) {
        const int row = m0 + i * 16 + v + half * 8;
        out[(size_t)row * P_SZ + n0 + j * 16 + r] = acc[i][j][v] + bias;
      }
    }
  }
}

// ---------------------------------------------------------------------
extern "C" void kernel_launch(void* const* d_in, const int* in_sizes, int n_in,
                              void* d_out, int out_size, void* d_ws, size_t ws_size,
                              hipStream_t stream) {
  (void)in_sizes; (void)n_in; (void)out_size; (void)ws_size;
  const int*   x           = (const int*)  d_in[0];
  const float* t           = (const float*)d_in[1];
  const float* sxy         = (const float*)d_in[2];
  // d_in[3] (y_t), d_in[4] (y_s) unused by reference
  const float* h0          = (const float*)d_in[5];
  const int*   active_user = (const int*)  d_in[6];
  const int*   length      = (const int*)  d_in[7];
  const float* enc         = (const float*)d_in[8];
  const float* user_enc    = (const float*)d_in[9];
  // d_in[10] trans_rows implicit (repeat(arange(P),16))
  const int*   tcols       = (const int*)  d_in[11];
  const float* tvals       = (const float*)d_in[12];
  // d_in[13] inter_rows implicit
  const int*   icols       = (const int*)  d_in[14];
  const float* ivals       = (const float*)d_in[15];
  const float* W_ih        = (const float*)d_in[16];
  const float* W_hh        = (const float*)d_in[17];
  const float* b_ih        = (const float*)d_in[18];
  const float* b_hh        = (const float*)d_in[19];
  const float* fc_W        = (const float*)d_in[20];
  const float* fc_b        = (const float*)d_in[21];
  float* out = (float*)d_out;

  // workspace carve-up (~107 MB)
  char* ws = (char*)d_ws;
  const size_t sz_seq = (size_t)S_LEN * B_SZ * H_SZ * sizeof(float); // 52.4 MB
  float* x_emb = (float*)(ws);                                   // [S*B, H]
  float* z     = (float*)(ws + sz_seq);                          // [S*B, H] -> becomes h sequence
  float* upref = (float*)(ws + 2 * sz_seq);                      // [B, H]
  float* wbuf  = (float*)(ws + 2 * sz_seq + (size_t)B_SZ * H_SZ * sizeof(float));       // [B, S]
  float* cat   = (float*)(ws + 2 * sz_seq + (size_t)B_SZ * (H_SZ + S_LEN) * sizeof(float)); // [B, 2H]

  const int THREADS = 256;               // 8 waves per block
  const int WPB = THREADS / 32;

  // 1) graph-conv embeddings per check-in: S*B waves
  k_xemb<<<(S_LEN * B_SZ) / WPB, THREADS, 0, stream>>>(x, enc, tcols, tvals, x_emb);
  // 2) user preference for active users: B waves
  k_upref<<<B_SZ / WPB, THREADS, 0, stream>>>(active_user, enc, icols, ivals, upref);
  // 3) attention weights: S*B waves
  k_weights<<<(S_LEN * B_SZ) / WPB, THREADS, 0, stream>>>(x_emb, upref, t, sxy, length, wbuf);
  // 4) input projection GEMM: S*B/16 waves, each 16x128
  k_zgemm<<<(S_LEN * B_SZ / 16) / WPB, THREADS, 0, stream>>>(x_emb, W_ih, b_ih, b_hh, z);
  // 5) sequential RNN: one launch per step, in-place on z
  for (int s = 0; s < S_LEN; ++s) {
    const float* hprev = (s == 0) ? h0 : (z + (size_t)(s - 1) * B_SZ * H_SZ);
    k_rnn_step<<<((B_SZ / 16) * (H_SZ / 16)) / WPB, THREADS, 0, stream>>>(
        hprev, W_hh, z + (size_t)s * B_SZ * H_SZ);
  }
  // 6) flashback pooling + concat
  k_pool<<<B_SZ, H_SZ, 0, stream>>>(z, wbuf, user_enc, active_user, cat);
  // 7) final FC GEMM: 2 M-blocks x 625 N-groups, 8 waves per block
  k_fc<<<2 * (P_SZ / FC_NB), THREADS, 0, stream>>>(cat, fc_W, fc_b, out);
}